// SoftHashTensorSketch_21818433864276
// MI455X (gfx1250) — compile-verified
//
#include <hip/hip_runtime.h>
#include <hip/hip_bf16.h>

typedef float v2f __attribute__((ext_vector_type(2)));
typedef float v8f __attribute__((ext_vector_type(8)));

#define D128   128
#define STEPS  8192
#define NLEV   8
#define NCHAR  4
#define BFRAG  (D128 * D128)   // floats per (c,q) fragment block = 16384

// ---------------- Phase 0a: softmax probs -> B-fragment-ordered circulants -------
// grid 32 blocks (bid = c*8+q), 128 threads.
// Fragment layout per (c,q): flat = ((kk*8 + nt)*32 + lane)*2 + e
//   k = kk*4 + (lane>>4)*2 + e ; n = nt*16 + (lane&15) ; value = p[(n-k) mod 128]
__global__ __launch_bounds__(128) void k_prep(const float* __restrict__ hashW,   // [8][128][4]
                                              const float* __restrict__ signW,   // [8][4]
                                              float* __restrict__ Bfrag,         // [4][8][16384]
                                              float* __restrict__ sgn)           // [8][4]
{
    const int bid = blockIdx.x;
    const int c = bid >> 3, q = bid & 7;
    const int j = threadIdx.x;
    __shared__ float sh[D128];
    __shared__ float p[D128];

    const float hv = hashW[(q * D128 + j) * NCHAR + c];
    sh[j] = hv; __syncthreads();
    for (int s = 64; s > 0; s >>= 1) { if (j < s) sh[j] = fmaxf(sh[j], sh[j + s]); __syncthreads(); }
    const float mx = sh[0]; __syncthreads();
    const float e = expf(hv - mx);
    sh[j] = e; __syncthreads();
    for (int s = 64; s > 0; s >>= 1) { if (j < s) sh[j] += sh[j + s]; __syncthreads(); }
    p[j] = e / sh[0]; __syncthreads();

    float* Bb = Bfrag + (size_t)bid * BFRAG;
    for (int it = 0; it < D128; ++it) {
        const int flat = it * D128 + j;
        const int e2 = flat & 1;
        const int l  = (flat >> 1) & 31;
        const int nt = (flat >> 6) & 7;
        const int kk = (flat >> 9) & 31;
        const int k  = kk * 4 + ((l >> 4) << 1) + e2;
        const int n  = nt * 16 + (l & 15);
        Bb[flat] = p[(n - k) & (D128 - 1)];
    }
    if (j == 0) sgn[q * NCHAR + c] = 1.0f / (1.0f + expf(-signW[q * NCHAR + c]));
}

// ---------------- Phase 0b: binomial coefficients (f64 -> f32) -------------------
__global__ __launch_bounds__(256) void k_coef(float* __restrict__ coefW, float* __restrict__ coefInv)
{
    const int i = blockIdx.x * 256 + threadIdx.x;
    if (i >= STEPS) return;
    for (int L = 1; L <= NLEV; ++L) {
        double cw = 0.0;
        if (i >= L - 1) {
            cw = 1.0;
            for (int t = 0; t < L - 1; ++t) cw *= (double)(i - t) / (double)(t + 1);
        }
        coefW[(L - 1) * STEPS + i] = (float)cw;
        double ci = 0.0;
        if (i + 1 >= L) {
            double v = 1.0;
            for (int t = 0; t < L; ++t) v *= (double)(i + 1 - t) / (double)(t + 1);
            ci = 1.0 / v;
        }
        coefInv[(L - 1) * STEPS + i] = (float)ci;
    }
}

// ---------------- Phase 0c: level-0 trajectory = e0 rows -------------------------
__global__ __launch_bounds__(256) void k_fill_e0(float* __restrict__ T0)
{
    const int e = blockIdx.x * 256 + threadIdx.x;
    if (e < STEPS * D128) T0[e] = ((e & (D128 - 1)) == 0) ? 1.0f : 0.0f;
}

// ---------------- Per-level GEMM with per-row circulant selection ---------------
// W[i][:] = C(i,L-1)*sigmoid(sign)* ( Traj[L-1][i-1] @ M_{c_i} )
// grid 512 blocks (16 steps), 256 threads = 8 waves; wave w owns N-tile w.
__global__ __launch_bounds__(256) void k_gemm(const float* __restrict__ prevT,   // [8192][128]
                                              const float* __restrict__ Bfrag,   // [4][8][16384]
                                              const float* __restrict__ coefW,   // [8][8192]
                                              const float* __restrict__ sgn,     // [8][4]
                                              const int*   __restrict__ seq,     // [8192]
                                              float* __restrict__ Wout,          // [8192][128]
                                              int L)
{
    __shared__ float lA[16][132];     // row stride 132 -> 32 distinct LDS bank pairs for b64 loads
    __shared__ int   lC[16];
    __shared__ float lW[16];

    const int i0 = blockIdx.x * 16;
    const int t  = threadIdx.x;
    const int q  = L - 1;

    if (t < 16) {
        const int i = i0 + t;
        const int c = seq[i];
        lC[t] = c;
        lW[t] = coefW[q * STEPS + i] * sgn[q * NCHAR + c];
    }
    __syncthreads();

    for (int e = t; e < 16 * D128; e += 256) {     // weighted, 1-step-shifted prev rows
        const int r = e >> 7, k = e & (D128 - 1);
        const int i = i0 + r;
        float v;
        if (i == 0) v = (L == 1) ? prevT[k] : 0.0f;   // T[0]_{-1}=e0 ; T[L>=1]_{-1}=0
        else        v = prevT[(size_t)(i - 1) * D128 + k];
        lA[r][k] = v * lW[r];
    }
    __syncthreads();

    const int wave = t >> 5, lane = t & 31;
    const int half2 = (lane >> 4) << 1, l16 = lane & 15;

    v8f acc0 = {}, acc1 = {}, acc2 = {}, acc3 = {};
    const int myc = lC[l16];
    const float m0 = (myc == 0) ? 1.0f : 0.0f;
    const float m1 = (myc == 1) ? 1.0f : 0.0f;
    const float m2 = (myc == 2) ? 1.0f : 0.0f;
    const float m3 = (myc == 3) ? 1.0f : 0.0f;
    const v2f* __restrict__ B0 = (const v2f*)(Bfrag + (size_t)(0 * NLEV + q) * BFRAG) + wave * 32 + lane;
    const v2f* __restrict__ B1 = (const v2f*)(Bfrag + (size_t)(1 * NLEV + q) * BFRAG) + wave * 32 + lane;
    const v2f* __restrict__ B2 = (const v2f*)(Bfrag + (size_t)(2 * NLEV + q) * BFRAG) + wave * 32 + lane;
    const v2f* __restrict__ B3 = (const v2f*)(Bfrag + (size_t)(3 * NLEV + q) * BFRAG) + wave * 32 + lane;

    #pragma unroll 4
    for (int kk = 0; kk < D128 / 4; ++kk) {
        const v2f av = *(const v2f*)&lA[l16][kk * 4 + half2];   // ds_load_b64, unconditional
        const size_t bi = (size_t)kk * 256;                      // float2 stride per kk
        acc0 = __builtin_amdgcn_wmma_f32_16x16x4_f32(false, av * m0, false, B0[bi], (short)0, acc0, false, false);
        acc1 = __builtin_amdgcn_wmma_f32_16x16x4_f32(false, av * m1, false, B1[bi], (short)0, acc1, false, false);
        acc2 = __builtin_amdgcn_wmma_f32_16x16x4_f32(false, av * m2, false, B2[bi], (short)0, acc2, false, false);
        acc3 = __builtin_amdgcn_wmma_f32_16x16x4_f32(false, av * m3, false, B3[bi], (short)0, acc3, false, false);
    }

    const v8f dsum = (acc0 + acc1) + (acc2 + acc3);
    #pragma unroll
    for (int r = 0; r < 8; ++r) {
        const int m = r + ((lane >> 4) ? 8 : 0);    // D 16x16: VGPR r -> M=r / r+8
        Wout[(size_t)(i0 + m) * D128 + wave * 16 + l16] = dsum[r];
    }
}

// ---------------- Blocked prefix scan over steps (per column), in place ----------
__global__ __launch_bounds__(128) void k_scan_part(const float* __restrict__ W, float* __restrict__ part)
{
    const int b = blockIdx.x, j = threadIdx.x;     // 64 x 128
    const float* base = W + (size_t)b * 128 * D128;
    float s = 0.0f;
    for (int r = 0; r < 128; ++r) s += base[r * D128 + j];
    part[b * D128 + j] = s;
}

__global__ __launch_bounds__(128) void k_scan_top(float* __restrict__ part)
{
    const int j = threadIdx.x;
    float run = 0.0f;
    for (int b = 0; b < 64; ++b) {
        const float v = part[b * D128 + j];
        part[b * D128 + j] = run;               // exclusive
        run += v;
    }
}

__global__ __launch_bounds__(128) void k_scan_apply(float* __restrict__ W,        // in-place -> Traj[L]
                                                    const float* __restrict__ part,
                                                    const float* __restrict__ coefInv,
                                                    int L)
{
    const int b = blockIdx.x, j = threadIdx.x;
    float run = part[b * D128 + j];
    float* base = W + (size_t)b * 128 * D128;
    const float* ci = coefInv + (L - 1) * STEPS + b * 128;
    for (int r = 0; r < 128; ++r) {
        run += base[r * D128 + j];
        base[r * D128 + j] = run * ci[r];       // T[L]_i = S_i / C(i+1, L)
    }
}

// ---------------- Output: Tp[P] (Tm is identically zero) -------------------------
__global__ __launch_bounds__(128) void k_out(const float* __restrict__ T, float* __restrict__ out)
{
    const int j = threadIdx.x;
    out[j] = T[(size_t)(STEPS - 1) * D128 + j];
}

extern "C" void kernel_launch(void* const* d_in, const int* in_sizes, int n_in,
                              void* d_out, int out_size, void* d_ws, size_t ws_size,
                              hipStream_t stream)
{
    (void)in_sizes; (void)n_in; (void)out_size; (void)ws_size;
    const int*   seq   = (const int*)d_in[0];     // (8192,) int32
    const float* hashW = (const float*)d_in[1];   // (8,128,4) f32
    const float* signW = (const float*)d_in[2];   // (8,4) f32
    float* out = (float*)d_out;                   // (128,) f32

    char* ws = (char*)d_ws;
    float* Bfrag   = (float*)(ws + 0);                       // 2 MB
    float* sgn     = (float*)(ws + (2u << 20));              // 128 B
    float* coefW   = (float*)(ws + (2u << 20) + 4096);       // 256 KB
    float* coefInv = (float*)(ws + (2u << 20) + 4096 + (256u << 10));  // 256 KB
    float* part    = (float*)(ws + (2u << 20) + 4096 + (512u << 10));  // 32 KB
    float* bufs[2] = { (float*)(ws + (4u << 20)),            // 4 MB each
                       (float*)(ws + (8u << 20)) };

    k_prep   <<<NCHAR * NLEV, 128, 0, stream>>>(hashW, signW, Bfrag, sgn);
    k_coef   <<<STEPS / 256, 256, 0, stream>>>(coefW, coefInv);
    k_fill_e0<<<(STEPS * D128) / 256, 256, 0, stream>>>(bufs[0]);

    for (int L = 1; L <= NLEV; ++L) {
        const float* prev = bufs[(L - 1) & 1];
        float*       cur  = bufs[L & 1];
        k_gemm      <<<STEPS / 16, 256, 0, stream>>>(prev, Bfrag, coefW, sgn, seq, cur, L);
        k_scan_part <<<64, 128, 0, stream>>>(cur, part);
        k_scan_top  <<<1, 128, 0, stream>>>(part);
        k_scan_apply<<<64, 128, 0, stream>>>(cur, part, coefInv, L);
    }
    k_out<<<1, 128, 0, stream>>>(bufs[NLEV & 1], out);
}